// EntityRepr_88132728914534
// MI455X (gfx1250) — compile-verified
//
#include <hip/hip_runtime.h>

typedef __attribute__((ext_vector_type(2))) float v2f;
typedef __attribute__((ext_vector_type(8))) float v8f;

// Problem constants from the reference
#define NB 32    // batch
#define NL 512   // seq len
#define NH 768   // hidden
#define NE 32    // entities
#define NM 8     // mentions / entity
#define NS 4     // span tokens / mention
#define NHC (NH / 16)  // 48 h-chunks of 16

__global__ __launch_bounds__(256) void EntityRepr_88132728914534_kernel(
    const float* __restrict__ tok,   // [B, L, H] f32
    const int*   __restrict__ idx,   // [B, E, M, S] int32
    float*       __restrict__ out)   // entity | mentions | mask, concatenated
{
    float* entity  = out;                                  // B*E*H   = 786432
    float* mention = out + (long)NB * NE * NH;             // B*E*M*H = 6291456
    float* mask    = out + (long)NB * NE * NH + (long)NB * NE * NM * NH;

    const int lane  = threadIdx.x & 31;
    const int row   = lane & 15;      // h offset within chunk (A-matrix row group)
    const int half  = lane >> 4;      // 0 -> span s=0,1 ; 1 -> span s=2,3

    const int wave  = blockIdx.x * (blockDim.x >> 5) + (threadIdx.x >> 5);
    // wave -> (b, e, h-chunk); total waves = NB*NE*NHC = 49152
    const int hc = wave % NHC;
    const int e  = (wave / NHC) % NE;
    const int b  = wave / (NHC * NE);

    const int  h0       = hc * 16;
    const long tok_b    = (long)b * NL * NH;
    const int  idx_base = ((b * NE + e) * NM) * NS;

    // B matrix: every element = 1/S. Constant fill makes the physical lane
    // layout of B irrelevant: D[M,N] = sum_s A[M,s] * 0.25 = span mean.
    const v2f bmat = {0.25f, 0.25f};

    v8f acc;
    #pragma unroll
    for (int j = 0; j < 8; ++j) acc[j] = 0.0f;

    #pragma unroll
    for (int m = 0; m < NM; ++m) {
        // Coalesced gather of A per documented 16x4 f32 A layout:
        //   lanes 0-15 : A[row, 0..1]  (span tokens s=0,1 at h0+row)
        //   lanes 16-31: A[row, 2..3]  (span tokens s=2,3 at h0+row)
        const int ib = idx_base + m * NS + half * 2;
        const int i0 = idx[ib + 0];
        const int i1 = idx[ib + 1];
        v2f a;
        a.x = tok[tok_b + (long)i0 * NH + h0 + row];
        a.y = tok[tok_b + (long)i1 * NH + h0 + row];

        // Prefetch next mention's gather targets (global_prefetch_b8) to
        // overlap random-index latency with the WMMA.
        if (m + 1 < NM) {
            const int pb = idx_base + (m + 1) * NS + half * 2;
            __builtin_prefetch(&tok[tok_b + (long)idx[pb + 0] * NH + h0 + row], 0, 3);
            __builtin_prefetch(&tok[tok_b + (long)idx[pb + 1] * NH + h0 + row], 0, 3);
        }

        v8f c;
        #pragma unroll
        for (int j = 0; j < 8; ++j) c[j] = 0.0f;

        // D = A(16x4) * B(4x16) + 0 ; f32 in, f32 out (no precision loss).
        v8f d = __builtin_amdgcn_wmma_f32_16x16x4_f32(
            /*neg_a=*/false, a, /*neg_b=*/false, bmat,
            /*c_mod=*/(short)0, c, /*reuse_a=*/false, /*reuse_b=*/false);

        // Entity accumulation: identical replicated D layout per mention,
        // so a plain per-lane vector add suffices (no cross-lane traffic).
        #pragma unroll
        for (int j = 0; j < 8; ++j) acc[j] += d[j];

        // Store mention means. D layout: VGPR j = row M=j (lanes 0-15),
        // M = j+8 (lanes 16-31); columns replicated. Lane 0 stores h0..h0+7,
        // lane 16 stores h0+8..h0+15; two 16B-aligned float4 stores each.
        if (row == 0) {
            float* p = mention + ((long)((b * NE + e) * NM + m)) * NH + h0 + half * 8;
            *(float4*)(p + 0) = make_float4(d[0], d[1], d[2], d[3]);
            *(float4*)(p + 4) = make_float4(d[4], d[5], d[6], d[7]);
        }
    }

    // Entity means: mean over the 8 mention means.
    if (row == 0) {
        const float s = 1.0f / (float)NM;
        float* p = entity + ((long)(b * NE + e)) * NH + h0 + half * 8;
        *(float4*)(p + 0) = make_float4(acc[0] * s, acc[1] * s, acc[2] * s, acc[3] * s);
        *(float4*)(p + 4) = make_float4(acc[4] * s, acc[5] * s, acc[6] * s, acc[7] * s);
    }

    // Mask: all ones, written by the first B*E*M global threads.
    const int gtid = blockIdx.x * blockDim.x + threadIdx.x;
    if (gtid < NB * NE * NM) mask[gtid] = 1.0f;
}

extern "C" void kernel_launch(void* const* d_in, const int* in_sizes, int n_in,
                              void* d_out, int out_size, void* d_ws, size_t ws_size,
                              hipStream_t stream) {
    const float* tok = (const float*)d_in[0];   // token_reprs f32 [32,512,768]
    const int*   idx = (const int*)d_in[1];     // pos_idx int32 [32,32,8,4]
    float*       out = (float*)d_out;

    // 49152 waves (one per (b,e,h-chunk)) = 6144 blocks of 256 threads (8 waves).
    dim3 grid(NB * NE * NHC / 8);
    dim3 block(256);
    hipLaunchKernelGGL(EntityRepr_88132728914534_kernel, grid, block, 0, stream,
                       tok, idx, out);
}